// weight_quantize_fn_log_19335942767060
// MI455X (gfx1250) — compile-verified
//
#include <hip/hip_runtime.h>
#include <math.h>

// Streaming power-quant for MI455X (gfx1250, wave32).
// Pass 1: absmax(tanh(x)) reduction -> alpha in d_ws (RT loads keep the
//         134MB weight resident in the 192MB L2).
// Pass 2: re-read (L2 hits), recompute tanh + power quantization, NT stores.
// Memory-bound: ~268MB HBM traffic -> ~11.5us floor at 23.3 TB/s.

typedef float v4f __attribute__((ext_vector_type(4)));

#define PB_F   0.872f
#define NLEV_F 255.0f

// Raw gfx1250 transcendental instructions.
__device__ __forceinline__ float fast_exp2(float x) { return __builtin_amdgcn_exp2f(x); }
__device__ __forceinline__ float fast_log2(float x) { return __builtin_amdgcn_logf(x); }

// CDNA5 hardware tanh (V_TANH_F32) if the toolchain exposes it; else ocml.
__device__ __forceinline__ float fast_tanh(float x) {
#if __has_builtin(__builtin_amdgcn_tanhf)
  return __builtin_amdgcn_tanhf(x);
#else
  return tanhf(x);
#endif
}

__device__ __forceinline__ float wave_max_f32(float v) {
  // wave32 butterfly reduction
  v = fmaxf(v, __shfl_xor(v, 16, 32));
  v = fmaxf(v, __shfl_xor(v,  8, 32));
  v = fmaxf(v, __shfl_xor(v,  4, 32));
  v = fmaxf(v, __shfl_xor(v,  2, 32));
  v = fmaxf(v, __shfl_xor(v,  1, 32));
  return v;
}

__global__ void pq_init_ws(unsigned int* ws) { ws[0] = 0u; }

__global__ __launch_bounds__(256)
void pq_pass1_absmax(const float* __restrict__ in,
                     unsigned int* __restrict__ ws,
                     long n4, long n) {
  const long stride = (long)gridDim.x * blockDim.x;
  const long gid    = (long)blockIdx.x * blockDim.x + threadIdx.x;

  float m = 0.0f;
  const v4f* in4 = (const v4f*)in;
  for (long i = gid; i < n4; i += stride) {
    v4f x = in4[i];                       // default RT hint: stay in L2 for pass 2
    m = fmaxf(m, fabsf(fast_tanh(x.x)));
    m = fmaxf(m, fabsf(fast_tanh(x.y)));
    m = fmaxf(m, fabsf(fast_tanh(x.z)));
    m = fmaxf(m, fabsf(fast_tanh(x.w)));
  }
  if (gid == 0) {                          // scalar tail (empty for this shape)
    for (long k = n4 * 4; k < n; ++k) m = fmaxf(m, fabsf(fast_tanh(in[k])));
  }

  __shared__ float sm[8];                  // 256 threads = 8 waves
  float w = wave_max_f32(m);
  const int lane = threadIdx.x & 31;
  const int wid  = threadIdx.x >> 5;
  if (lane == 0) sm[wid] = w;
  __syncthreads();
  if (wid == 0) {
    float v = (lane < (int)(blockDim.x >> 5)) ? sm[lane] : 0.0f;
    v = wave_max_f32(v);
    // all values >= 0: uint ordering == float ordering
    if (lane == 0) atomicMax(ws, __float_as_uint(v));
  }
}

__device__ __forceinline__ float pq_quant(float x, float alpha, float inv_alpha,
                                          float ref, float scale, float inv_scale,
                                          float log2pb, float inv_log2pb) {
  float w   = fast_tanh(x);
  float sgn = (w > 0.0f) ? 1.0f : ((w < 0.0f) ? -1.0f : 0.0f);
  float xs  = fabsf(w) * inv_alpha * scale;
  float t   = xs + ref;
  float lv  = fast_log2(t) * inv_log2pb;             // log(t)/log(pb)
  float ll  = fminf(fmaxf(floorf(lv), 0.0f), NLEV_F);
  float lu  = fminf(fmaxf(ceilf(lv),  0.0f), NLEV_F);
  float xql = fast_exp2(ll * log2pb);                // pb^ll
  float xqu = fast_exp2(lu * log2pb);                // pb^lu
  xql = (xql < ref) ? ref  : xql;
  xqu = (xqu < ref) ? 0.0f : xqu;
  float xq  = (fabsf(t - xqu) < fabsf(t - xql)) ? xqu : xql;  // ties -> lower
  return (xq - ref) * inv_scale * sgn * alpha;
}

__global__ __launch_bounds__(256)
void pq_pass2_quant(const float* __restrict__ in,
                    float* __restrict__ out,
                    const unsigned int* __restrict__ ws,
                    long n4, long n) {
  const float alpha      = __uint_as_float(ws[0]);   // uniform -> scalar load
  const float inv_alpha  = 1.0f / alpha;
  const float log2pb     = fast_log2(PB_F);
  const float ref        = fast_exp2(NLEV_F * log2pb); // pb^255 (~6.79e-16)
  const float scale      = 1.0f - ref;
  const float inv_scale  = 1.0f / scale;
  const float inv_log2pb = 1.0f / log2pb;

  const long i = (long)blockIdx.x * blockDim.x + threadIdx.x;
  if (i < n4) {
    v4f x = ((const v4f*)in)[i];                     // expected L2 hit
    v4f r;
    r.x = pq_quant(x.x, alpha, inv_alpha, ref, scale, inv_scale, log2pb, inv_log2pb);
    r.y = pq_quant(x.y, alpha, inv_alpha, ref, scale, inv_scale, log2pb, inv_log2pb);
    r.z = pq_quant(x.z, alpha, inv_alpha, ref, scale, inv_scale, log2pb, inv_log2pb);
    r.w = pq_quant(x.w, alpha, inv_alpha, ref, scale, inv_scale, log2pb, inv_log2pb);
    // write-once output: non-temporal store, don't pollute L2
    __builtin_nontemporal_store(r, &((v4f*)out)[i]);
  }
  if (i == 0) {                                      // scalar tail (empty here)
    for (long k = n4 * 4; k < n; ++k)
      out[k] = pq_quant(in[k], alpha, inv_alpha, ref, scale, inv_scale,
                        log2pb, inv_log2pb);
  }
}

extern "C" void kernel_launch(void* const* d_in, const int* in_sizes, int n_in,
                              void* d_out, int out_size, void* d_ws, size_t ws_size,
                              hipStream_t stream) {
  (void)n_in; (void)out_size; (void)ws_size;
  const float*  in  = (const float*)d_in[0];
  float*        out = (float*)d_out;
  unsigned int* ws  = (unsigned int*)d_ws;

  const long n  = (long)in_sizes[0];
  const long n4 = n >> 2;

  // deterministic: reset the reduction cell every launch
  pq_init_ws<<<1, 1, 0, stream>>>(ws);

  long b1 = (n4 + 255) / 256;
  if (b1 > 8192) b1 = 8192;          // grid-stride; ~8K atomics total
  if (b1 < 1)    b1 = 1;
  pq_pass1_absmax<<<(int)b1, 256, 0, stream>>>(in, ws, n4, n);

  long b2 = (n4 + 255) / 256;        // one float4 per thread
  if (b2 < 1) b2 = 1;
  pq_pass2_quant<<<(int)b2, 256, 0, stream>>>(in, out, ws, n4, n);
}